// AT_GT_80401787781737
// MI455X (gfx1250) — compile-verified
//
#include <hip/hip_runtime.h>
#include <stdint.h>

// Image geometry (reference: (1, 4096, 4096, 3) float32, NHWC).
#define H_IMG 4096
#define W_IMG 4096
#define C_IMG 3
#define WF    (W_IMG * C_IMG)      // 12288 floats per image row (flattened W*C)

// Tile geometry. Horizontal taps are at flat offsets +-3 (channel stride),
// so a 4-float halo keeps everything 16-byte aligned.
#define TX    512                  // tile width in floats (divides WF: 24 tiles/row)
#define TY    16                   // tile height in rows  (divides H: 256 tiles/col)
#define HX    4                    // x halo (need 3, pad to 4 for b128 alignment)
#define SW    (TX + 2 * HX)        // 520 floats per LDS row
#define SH    (TY + 2)             // 18 LDS rows (1-row y halo each side)
#define NLD   ((SW / 4) * SH)      // 2340 16-byte chunks to stage
#define NST   ((TX / 4) * TY)      // 2048 16-byte output chunks
#define NTHR  256                  // 8 waves (wave32)

// Native clang vector: usable with __builtin_nontemporal_store (HIP's float4
// is a class type and is rejected by the builtin).
typedef float v4f __attribute__((ext_vector_type(4)));

__device__ __forceinline__ float inv_anscombe(float y) {
    // clip then unbiased closed-form inverse Anscombe
    y = fminf(fmaxf(y, 0.0f), 255.0f);
    const float r  = 1.0f / y;
    const float r2 = r * r;
    // 0.25*sqrt(1.5) = 0.30618621784789724 ; 0.625*sqrt(1.5) = 0.7654655446197431
    return 0.25f * y * y - 0.125f
         + 0.30618621784789724f * r
         - 1.375f * r2
         + 0.76546554461974310f * r2 * r;
}

__global__ __launch_bounds__(NTHR)
void AT_GT_anscombe_blur_kernel(const float* __restrict__ src, float* __restrict__ dst) {
    __shared__ float tile[SW * SH];

    const int tid  = threadIdx.x;
    const int row0 = blockIdx.y * TY;   // first output row of this tile
    const int col0 = blockIdx.x * TX;   // first output flat-col of this tile

    // ---- Stage 1: async DMA (global -> LDS) of raw tile incl. halo ----------
    // Every 16B chunk is either fully in-bounds or fully out (TX | WF, 4 | HX),
    // OOB chunks get zeros (reference zero-pads the *transformed* image, and we
    // skip the transform for them in stage 2, so zero stays zero).
    for (int i = tid; i < NLD; i += NTHR) {
        const int sr   = i / (SW / 4);
        const int sc   = (i - sr * (SW / 4)) * 4;
        const int grow = row0 - 1 + sr;
        const int gcol = col0 - HX + sc;
        float* lp = &tile[sr * SW + sc];
        if ((unsigned)grow < (unsigned)H_IMG && (unsigned)gcol < (unsigned)WF) {
            const uint32_t lofs = (uint32_t)(uintptr_t)lp;      // low 32 bits = LDS byte addr
            const uint64_t ga   = (uint64_t)(const void*)(src + (size_t)grow * WF + gcol);
            // CDNA5 async copy, 16B per lane, tracked by ASYNCcnt (no VGPR staging).
            // Default RT temporal hint: halo rows are re-read by the vertically
            // adjacent tile, so we WANT these lines resident in the 192MB L2.
            asm volatile("global_load_async_to_lds_b128 %0, %1, off"
                         :: "v"(lofs), "v"(ga) : "memory");
        } else {
            lp[0] = 0.0f; lp[1] = 0.0f; lp[2] = 0.0f; lp[3] = 0.0f;
        }
    }
    asm volatile("s_wait_asynccnt 0" ::: "memory");
    __syncthreads();

    // ---- Stage 2: Anscombe transform in place (in-bounds entries only) ------
    for (int i = tid; i < NLD; i += NTHR) {
        const int sr   = i / (SW / 4);
        const int sc   = (i - sr * (SW / 4)) * 4;
        const int grow = row0 - 1 + sr;
        const int gcol = col0 - HX + sc;
        if ((unsigned)grow < (unsigned)H_IMG && (unsigned)gcol < (unsigned)WF) {
            v4f* lp = (v4f*)&tile[sr * SW + sc];
            v4f v = *lp;
            v.x = 2.0f * sqrtf(v.x + 0.375f);
            v.y = 2.0f * sqrtf(v.y + 0.375f);
            v.z = 2.0f * sqrtf(v.z + 0.375f);
            v.w = 2.0f * sqrtf(v.w + 0.375f);
            *lp = v;
        }
    }
    __syncthreads();

    // ---- Stage 3: 9-tap blur + clip + inverse Anscombe, coalesced stores ----
    // Separable Gaussian weights, sigma = 1.3 (matches reference normalization).
    const float t1  = expf(-1.0f / 3.38f);           // exp(-1/(2*sigma^2))
    const float t2  = t1 * t1;                       // exp(-2/(2*sigma^2))
    const float nrm = 1.0f / (1.0f + 4.0f * t1 + 4.0f * t2);
    const float kc = nrm;        // center
    const float ke = t1 * nrm;   // edge
    const float kk = t2 * nrm;   // corner

    for (int j = tid; j < NST; j += NTHR) {
        const int orow = j >> 7;            // 0..TY-1   (TX/4 == 128)
        const int oc   = (j & 127) * 4;     // 0..TX-4

        float ax = 0.0f, ay = 0.0f, az = 0.0f, aw = 0.0f;
        #pragma unroll
        for (int r = 0; r < 3; ++r) {
            const float wl = (r == 1) ? ke : kk;   // taps at x-3 / x+3
            const float wm = (r == 1) ? kc : ke;   // tap at x
            const float* rp = &tile[(orow + r) * SW + HX + oc];
            const v4f A  = *(const v4f*)(rp - 4);
            const v4f B  = *(const v4f*)(rp);
            const v4f Cv = *(const v4f*)(rp + 4);
            // lane p needs in[p-3], in[p], in[p+3] (flat channel-preserving taps)
            ax += wl * A.y + wm * B.x + wl * B.w;
            ay += wl * A.z + wm * B.y + wl * Cv.x;
            az += wl * A.w + wm * B.z + wl * Cv.y;
            aw += wl * B.x + wm * B.w + wl * Cv.z;
        }

        v4f o;
        o.x = inv_anscombe(ax);
        o.y = inv_anscombe(ay);
        o.z = inv_anscombe(az);
        o.w = inv_anscombe(aw);
        // Output is write-once / never re-read: non-temporal store keeps the
        // 192MB L2 free for input lines that still have cross-tile reuse.
        __builtin_nontemporal_store(o, (v4f*)(dst + (size_t)(row0 + orow) * WF + col0 + oc));
    }
}

extern "C" void kernel_launch(void* const* d_in, const int* in_sizes, int n_in,
                              void* d_out, int out_size, void* d_ws, size_t ws_size,
                              hipStream_t stream) {
    (void)in_sizes; (void)n_in; (void)out_size; (void)d_ws; (void)ws_size;
    const float* src = (const float*)d_in[0];
    float*       dst = (float*)d_out;
    dim3 grid(WF / TX, H_IMG / TY);   // 24 x 256 = 6144 blocks
    dim3 block(NTHR);
    AT_GT_anscombe_blur_kernel<<<grid, block, 0, stream>>>(src, dst);
}